// CyclicVQ_40046275068125
// MI455X (gfx1250) — compile-verified
//
#include <hip/hip_runtime.h>
#include <stdint.h>

typedef float v4f __attribute__((ext_vector_type(4)));
typedef int   v4i __attribute__((ext_vector_type(4)));

#define PI_F      3.14159265358979323846f
#define TWO_PI_F  6.28318530717958647692f

#define THREADS          256
#define TILES_PER_BLOCK  4
// per-thread per-tile: 12 floats angles (48B) + 8 ints mask (32B)
#define ANG_BYTES_PER_TILE (THREADS * 48)   // 12288
#define MSK_BYTES_PER_TILE (THREADS * 32)   // 8192
#define BUF_BYTES (ANG_BYTES_PER_TILE + MSK_BYTES_PER_TILE)  // 20480

// bin params: n = {24, 12, 16}
#define INV0 (24.0f / TWO_PI_F)
#define INV1 (12.0f / TWO_PI_F)
#define INV2 (16.0f / TWO_PI_F)
#define W0   (TWO_PI_F / 24.0f)
#define W1   (TWO_PI_F / 12.0f)
#define W2   (TWO_PI_F / 16.0f)

__device__ __forceinline__ void quant1(float a, float inv, float width, int nmax,
                                       float& q, int& idx) {
    // a >= -pi  =>  (a+pi)*inv >= 0, so int-cast truncation == floor.
    // Clamp absorbs any -epsilon rounding at the lower edge.
    int k = (int)((a + PI_F) * inv);
    k = k < 0 ? 0 : (k > nmax ? nmax : k);
    idx = k;
    q = fmaf(width, (float)k + 0.5f, -PI_F);   // center = -pi + width*(k+0.5)
}

__device__ __forceinline__ void process4(const float* a12, const int* m8,
                                         float* qres, float* ires) {
#pragma unroll
    for (int p = 0; p < 4; ++p) {
        float q0, q1, q2; int i0, i1, i2;
        quant1(a12[3 * p + 0], INV0, W0, 23, q0, i0);
        quant1(a12[3 * p + 1], INV1, W1, 11, q1, i1);
        quant1(a12[3 * p + 2], INV2, W2, 15, q2, i2);
        if (m8[2 * p + 0] != 0) { q0 = 0.0f; i0 = 24; }  // NULL trans
        if (m8[2 * p + 1] != 0) { q1 = 0.0f; i1 = 12; }  // NULL rot
        qres[3 * p + 0] = q0; qres[3 * p + 1] = q1; qres[3 * p + 2] = q2;
        ires[3 * p + 0] = (float)i0; ires[3 * p + 1] = (float)i1; ires[3 * p + 2] = (float)i2;
    }
}

// Issue 5 async b128 copies (48B angles + 32B mask for this lane's 4 pixels)
// into this lane's private LDS staging slot, using the SADDR (GVS) form:
// mem_addr = SGPR_base + VGPR_offset32 + inst_offset. ASYNCcnt += 5.
__device__ __forceinline__ void issue_tile(uint32_t la, uint32_t lm,
                                           uint64_t abase, uint64_t mbase,
                                           uint32_t aoff, uint32_t moff) {
    asm volatile(
        "global_load_async_to_lds_b128 %0, %4, %2\n\t"
        "global_load_async_to_lds_b128 %0, %4, %2 offset:16\n\t"
        "global_load_async_to_lds_b128 %0, %4, %2 offset:32\n\t"
        "global_load_async_to_lds_b128 %1, %5, %3\n\t"
        "global_load_async_to_lds_b128 %1, %5, %3 offset:16"
        :: "v"(la), "v"(lm), "s"(abase), "s"(mbase), "v"(aoff), "v"(moff)
        : "memory");
}

__global__ __launch_bounds__(THREADS, 1) void cyclicvq_main(
    const float* __restrict__ angles, const int* __restrict__ mask,
    float* __restrict__ qout, float* __restrict__ iout) {
    __shared__ __align__(16) char smem[2][BUF_BYTES];

    const int tid = threadIdx.x;
    const int block_base = (int)blockIdx.x * (THREADS * TILES_PER_BLOCK);

    // lane-private LDS slots (low 32 bits of flat shared addr = LDS byte offset)
    const uint32_t la0 = (uint32_t)(uintptr_t)(&smem[0][0]) + (uint32_t)tid * 48u;
    const uint32_t la1 = (uint32_t)(uintptr_t)(&smem[1][0]) + (uint32_t)tid * 48u;
    const uint32_t lm0 = (uint32_t)(uintptr_t)(&smem[0][ANG_BYTES_PER_TILE]) + (uint32_t)tid * 32u;
    const uint32_t lm1 = (uint32_t)(uintptr_t)(&smem[1][ANG_BYTES_PER_TILE]) + (uint32_t)tid * 32u;
    const uint32_t la[2] = { la0, la1 };
    const uint32_t lm[2] = { lm0, lm1 };

    const uint64_t abase = (uint64_t)(uintptr_t)angles;
    const uint64_t mbase = (uint64_t)(uintptr_t)mask;
    const uint32_t aoff0 = (uint32_t)(block_base + tid) * 48u;  // tile-0 byte offsets
    const uint32_t moff0 = (uint32_t)(block_base + tid) * 32u;

    // prime the pipeline with tile 0
    issue_tile(la[0], lm[0], abase, mbase, aoff0, moff0);

#pragma unroll
    for (int t = 0; t < TILES_PER_BLOCK; ++t) {
        if (t + 1 < TILES_PER_BLOCK) {
            // prefetch next tile into the other buffer (32-bit immediate adds only)
            issue_tile(la[(t + 1) & 1], lm[(t + 1) & 1], abase, mbase,
                       aoff0 + (uint32_t)(t + 1) * ANG_BYTES_PER_TILE,
                       moff0 + (uint32_t)(t + 1) * MSK_BYTES_PER_TILE);
            asm volatile("s_wait_asynccnt 5" ::: "memory");  // current tile's 5 done (in-order)
        } else {
            asm volatile("s_wait_asynccnt 0" ::: "memory");
        }

        const v4f* ap = (const v4f*)(&smem[t & 1][0] + tid * 48);
        const v4i* mp = (const v4i*)(&smem[t & 1][ANG_BYTES_PER_TILE] + tid * 32);
        v4f A0 = ap[0], A1 = ap[1], A2 = ap[2];      // ds_load_b128 x3
        v4i M0 = mp[0], M1 = mp[1];                  // ds_load_b128 x2

        float a12[12] = { A0.x, A0.y, A0.z, A0.w,
                          A1.x, A1.y, A1.z, A1.w,
                          A2.x, A2.y, A2.z, A2.w };
        int   m8[8]   = { M0.x, M0.y, M0.z, M0.w, M1.x, M1.y, M1.z, M1.w };

        v4f qv[3], iv[3];
        process4(a12, m8, (float*)qv, (float*)iv);

        // 32-bit byte offsets so stores use the saddr + zext(voffset) form
        const uint32_t ooff = (uint32_t)(block_base + t * THREADS + tid) * 48u;
        v4f* qp = (v4f*)((char*)qout + (uint64_t)ooff);
        v4f* ip = (v4f*)((char*)iout + (uint64_t)ooff);
        __builtin_nontemporal_store(qv[0], qp + 0);   // outputs are never re-read
        __builtin_nontemporal_store(qv[1], qp + 1);
        __builtin_nontemporal_store(qv[2], qp + 2);
        __builtin_nontemporal_store(iv[0], ip + 0);
        __builtin_nontemporal_store(iv[1], ip + 1);
        __builtin_nontemporal_store(iv[2], ip + 2);
    }
}

// scalar tail for any pixels not covered by the tiled kernel
__global__ void cyclicvq_tail(const float* __restrict__ angles,
                              const int* __restrict__ mask,
                              float* __restrict__ qout, float* __restrict__ iout,
                              int start_pix, int npix) {
    int p = start_pix + (int)blockIdx.x * blockDim.x + (int)threadIdx.x;
    if (p >= npix) return;
    float a0 = angles[(size_t)p * 3 + 0];
    float a1 = angles[(size_t)p * 3 + 1];
    float a2 = angles[(size_t)p * 3 + 2];
    float q0, q1, q2; int i0, i1, i2;
    quant1(a0, INV0, W0, 23, q0, i0);
    quant1(a1, INV1, W1, 11, q1, i1);
    quant1(a2, INV2, W2, 15, q2, i2);
    if (mask[(size_t)p * 2 + 0] != 0) { q0 = 0.0f; i0 = 24; }
    if (mask[(size_t)p * 2 + 1] != 0) { q1 = 0.0f; i1 = 12; }
    qout[(size_t)p * 3 + 0] = q0; qout[(size_t)p * 3 + 1] = q1; qout[(size_t)p * 3 + 2] = q2;
    iout[(size_t)p * 3 + 0] = (float)i0; iout[(size_t)p * 3 + 1] = (float)i1;
    iout[(size_t)p * 3 + 2] = (float)i2;
}

extern "C" void kernel_launch(void* const* d_in, const int* in_sizes, int n_in,
                              void* d_out, int out_size, void* d_ws, size_t ws_size,
                              hipStream_t stream) {
    const float* angles = (const float*)d_in[0];   // (4096*2048, 3) fp32
    const int*   mask   = (const int*)d_in[1];     // (4096*2048, 2) int
    float* qout = (float*)d_out;                   // quantized, in_sizes[0] floats
    float* iout = qout + (size_t)in_sizes[0];      // indices (as fp32 values)

    const int total_slots = in_sizes[0] / 12;              // 4 pixels per slot
    const int slots_per_block = THREADS * TILES_PER_BLOCK; // 1024
    const int blocks = total_slots / slots_per_block;      // 2048 for 4096x2048

    if (blocks > 0) {
        cyclicvq_main<<<blocks, THREADS, 0, stream>>>(angles, mask, qout, iout);
    }

    const int npix = in_sizes[0] / 3;
    const int done_pix = blocks * slots_per_block * 4;
    const int rem = npix - done_pix;
    if (rem > 0) {
        cyclicvq_tail<<<(rem + 255) / 256, 256, 0, stream>>>(angles, mask, qout, iout,
                                                             done_pix, npix);
    }
}